// LightGCN_18150531793440
// MI455X (gfx1250) — compile-verified
//
#include <hip/hip_runtime.h>
#include <hip/hip_bf16.h>

#define N_USERS  100000
#define N_ITEMS  200000
#define N_TOTAL  300000
#define DIM      64
#define BATCH    16384

typedef __attribute__((ext_vector_type(2))) float v2f;
typedef __attribute__((ext_vector_type(8))) float v8f;

// ---------------------------------------------------------------------------
// bufA = concat(user_emb, item_emb); acc = same (layer-0 term of the mean)
// ---------------------------------------------------------------------------
__global__ void lg_init_concat(const float* __restrict__ uemb,
                               const float* __restrict__ iemb,
                               float* __restrict__ bufA,
                               float* __restrict__ acc) {
    const int i4 = blockIdx.x * blockDim.x + threadIdx.x;   // float4 index
    const int i  = i4 << 2;
    const int nU = N_USERS * DIM;                           // divisible by 4
    if (i >= N_TOTAL * DIM) return;
    const float4 x = (i < nU) ? *(const float4*)(uemb + i)
                              : *(const float4*)(iemb + (i - nU));
    *(float4*)(bufA + i) = x;
    *(float4*)(acc  + i) = x;
}

__global__ void lg_zero(float* __restrict__ dst, int n) {
    const int i = (blockIdx.x * blockDim.x + threadIdx.x) << 2;
    if (i < n) *(float4*)(dst + i) = make_float4(0.f, 0.f, 0.f, 0.f);
}

__global__ void lg_accum(float* __restrict__ acc, const float* __restrict__ src, int n) {
    const int i = (blockIdx.x * blockDim.x + threadIdx.x) << 2;
    if (i >= n) return;
    float4 a = *(const float4*)(acc + i);
    const float4 s = *(const float4*)(src + i);
    a.x += s.x; a.y += s.y; a.z += s.z; a.w += s.w;
    *(float4*)(acc + i) = a;
}

// ---------------------------------------------------------------------------
// SpMM scatter: out[row] += val * in[col], 64-dim rows.
// One wave = 32 edges. Meta loaded coalesced (edge-per-lane, non-temporal so
// the edge streams don't evict the L2-resident feature tables), broadcast via
// wave32 shfl. Two edges in flight per wave: 16 lanes x float4 per edge.
// ---------------------------------------------------------------------------
__global__ void lg_spmm(const float* __restrict__ in, float* __restrict__ out,
                        const int* __restrict__ erow, const int* __restrict__ ecol,
                        const float* __restrict__ eval, int nEdges) {
    const int lane = threadIdx.x & 31;
    const int wave = blockIdx.x * (blockDim.x >> 5) + (threadIdx.x >> 5);
    const int edgeBase = wave << 5;
    if (edgeBase >= nEdges) return;

    const int e = edgeBase + lane;
    int r = 0, c = 0; float v = 0.f;
    if (e < nEdges) {
        r = __builtin_nontemporal_load(erow + e);
        c = __builtin_nontemporal_load(ecol + e);
        v = __builtin_nontemporal_load(eval + e);
        // stream-ahead into GL2 (global_prefetch_b8); speculative, OOB is safe
        __builtin_prefetch(erow + e + 4096, 0, 0);
        __builtin_prefetch(ecol + e + 4096, 0, 0);
        __builtin_prefetch(eval + e + 4096, 0, 0);
    }

    const int g = lane >> 4;            // which of 2 concurrent edges
    const int d = (lane & 15) << 2;     // float4 dim offset within the row
#pragma unroll
    for (int j = 0; j < 32; j += 2) {
        const int   er = __shfl(r, j + g, 32);
        const int   ec = __shfl(c, j + g, 32);
        const float ev = __shfl(v, j + g, 32);
        if (edgeBase + j + g < nEdges) {
            const float4 x = *(const float4*)(in + ec * DIM + d);
            float* o = out + er * DIM + d;
            atomicAdd(o + 0, ev * x.x);
            atomicAdd(o + 1, ev * x.y);
            atomicAdd(o + 2, ev * x.z);
            atomicAdd(o + 3, ev * x.w);
        }
    }
}

// ---------------------------------------------------------------------------
// Final stage: gamma[b] = dot(light_out[users[b]], light_out[N_USERS+items[b]])
// with light_out = acc/4. One wave per 16-batch tile:
//   C(16x16) = U(16x64) x V^T(64x16) via 16 chained V_WMMA_F32_16X16X4_F32,
//   gamma = diag(C) / 16.
// f32 A layout (ISA 7.12.2): lane l holds A[l&15][(l>>4)*2 .. +1] in 2 VGPRs.
// f32 B layout:              lane l holds B[(l>>4)*2 .. +1][l&15].
// Both reduce to a float2 load from row (l&15) at column k0+(l>>4)*2.
// C layout: VGPR j = M rows {j, j+8}; diag m sits in VGPR (m&7) of lane m
// (m<8) or lane m+16 (m>=8).
// ---------------------------------------------------------------------------
__global__ void lg_dot_wmma(const float* __restrict__ acc,
                            const int* __restrict__ users,
                            const int* __restrict__ items,
                            float* __restrict__ out) {
    const int lane = threadIdx.x & 31;
    const int tile = blockIdx.x * (blockDim.x >> 5) + (threadIdx.x >> 5); // 1024 tiles
    const int m  = lane & 15;
    const int kg = (lane >> 4) << 1;
    const int b  = tile * 16 + m;

    const int uRow = users[b] * DIM;
    const int vRow = (N_USERS + items[b]) * DIM;

    v8f c = {};
#pragma unroll
    for (int k0 = 0; k0 < DIM; k0 += 4) {
        const v2f a  = *(const v2f*)(acc + uRow + k0 + kg);
        const v2f bb = *(const v2f*)(acc + vRow + k0 + kg);
        c = __builtin_amdgcn_wmma_f32_16x16x4_f32(false, a, false, bb,
                                                  (short)0, c, false, false);
    }

    // diagonal extraction: wanted VGPR index is m&7 on lanes 0-7 and 24-31
    const int s = m & 7;
    float dv;
    if      (s == 0) dv = c[0];
    else if (s == 1) dv = c[1];
    else if (s == 2) dv = c[2];
    else if (s == 3) dv = c[3];
    else if (s == 4) dv = c[4];
    else if (s == 5) dv = c[5];
    else if (s == 6) dv = c[6];
    else             dv = c[7];

    if (lane < 8 || lane >= 24)
        out[tile * 16 + m] = dv * 0.0625f;   // (1/4 mean)^2
}

// ---------------------------------------------------------------------------
extern "C" void kernel_launch(void* const* d_in, const int* in_sizes, int n_in,
                              void* d_out, int out_size, void* d_ws, size_t ws_size,
                              hipStream_t stream) {
    const float* user_emb = (const float*)d_in[0];
    const float* item_emb = (const float*)d_in[1];
    const float* edge_val = (const float*)d_in[2];
    const int*   edge_row = (const int*)  d_in[3];
    const int*   edge_col = (const int*)  d_in[4];
    const int*   users    = (const int*)  d_in[5];
    const int*   items    = (const int*)  d_in[6];
    float*       out      = (float*)d_out;

    const int nEdges = in_sizes[3];
    const int nElem  = N_TOTAL * DIM;               // 19.2M floats per buffer

    float* bufA = (float*)d_ws;
    float* bufB = bufA + (size_t)nElem;
    float* acc  = bufB + (size_t)nElem;

    const int TPB   = 256;
    const int gElem = (nElem / 4 + TPB - 1) / TPB;                 // float4 grids
    const int gSpmm = (nEdges + 32 * (TPB / 32) - 1) / (32 * (TPB / 32));
    const int gDot  = (BATCH / 16) / (TPB / 32);                   // 1024 waves

    // acc = emb ; bufA = emb
    lg_init_concat<<<gElem, TPB, 0, stream>>>(user_emb, item_emb, bufA, acc);

    // e1 = S*emb  (NOT accumulated — reference skips layer 1)
    lg_zero<<<gElem, TPB, 0, stream>>>(bufB, nElem);
    lg_spmm<<<gSpmm, TPB, 0, stream>>>(bufA, bufB, edge_row, edge_col, edge_val, nEdges);

    // e2 = S*e1 ; acc += e2
    lg_zero<<<gElem, TPB, 0, stream>>>(bufA, nElem);
    lg_spmm<<<gSpmm, TPB, 0, stream>>>(bufB, bufA, edge_row, edge_col, edge_val, nEdges);
    lg_accum<<<gElem, TPB, 0, stream>>>(acc, bufA, nElem);

    // e3 = S*e2 ; acc += e3
    lg_zero<<<gElem, TPB, 0, stream>>>(bufB, nElem);
    lg_spmm<<<gSpmm, TPB, 0, stream>>>(bufA, bufB, edge_row, edge_col, edge_val, nEdges);
    lg_accum<<<gElem, TPB, 0, stream>>>(acc, bufB, nElem);

    // e4 = S*e3 ; acc += e4
    lg_zero<<<gElem, TPB, 0, stream>>>(bufA, nElem);
    lg_spmm<<<gSpmm, TPB, 0, stream>>>(bufB, bufA, edge_row, edge_col, edge_val, nEdges);
    lg_accum<<<gElem, TPB, 0, stream>>>(acc, bufA, nElem);

    // gamma = diag(U V^T) / 16 via f32 WMMA
    lg_dot_wmma<<<gDot, TPB, 0, stream>>>(acc, users, items, out);
}